// SegVoxHead_74380243632774
// MI455X (gfx1250) — compile-verified
//
#include <hip/hip_runtime.h>
#include <math.h>

// Problem constants (match reference)
#define BB 8
#define NN 16384
#define KK 2048
#define SS 64
#define CC 512
#define HH 200
#define WW 176

#define FPS_THREADS 1024
#define FPS_WAVES   (FPS_THREADS / 32)
#define PPT         (NN / FPS_THREADS)   // 16 points per thread

// ---------------------------------------------------------------------------
// float atomic max via sign-aware integer atomics (valid for -inf init,
// negative values, matches IEEE ordering for non-NaN data)
// ---------------------------------------------------------------------------
__device__ __forceinline__ void atomicMaxF(float* addr, float v) {
    if (v >= 0.0f) {
        atomicMax((int*)addr, __float_as_int(v));
    } else {
        atomicMin((unsigned int*)addr, __float_as_uint(v));
    }
}

// ---------------------------------------------------------------------------
// init output to -inf (segment_max identity for empty segments)
// ---------------------------------------------------------------------------
__global__ void init_out_kernel(float* __restrict__ out, int n) {
    int i = blockIdx.x * blockDim.x + threadIdx.x;
    if (i < n) out[i] = -__builtin_inff();
}

// ---------------------------------------------------------------------------
// Farthest point sampling: one workgroup per batch.
// Entire 192KB point cloud staged into LDS with gfx1250 async global->LDS DMA
// (ASYNCcnt path), per-thread point slice + running min-dist kept in VGPRs,
// wave32 shuffle argmax reduction + 32-entry LDS stage across waves.
// Distance uses explicit non-fused mul/add to replicate the reference's
// ((dx^2 + dy^2) + dz^2) rounding order exactly (argmax decisions cascade).
// ---------------------------------------------------------------------------
__global__ __launch_bounds__(FPS_THREADS)
void fps_kernel(const float* __restrict__ points, int* __restrict__ idxs) {
    extern __shared__ char smem[];
    float* pts = (float*)smem;                         // NN*3 floats (192 KB)
    float* rv  = (float*)(smem + (size_t)NN * 3 * 4);  // FPS_WAVES floats
    int*   ri  = (int*)(rv + FPS_WAVES);               // FPS_WAVES ints
    int*   win = ri + FPS_WAVES;                       // 1 int

    const int b = blockIdx.x;
    const int t = threadIdx.x;
    const float* gpts = points + (size_t)b * NN * 3;

    // ---- stage batch's points into LDS via async global->LDS -------------
#if __HIP_DEVICE_COMPILE__
    {
        const int chunks = (NN * 3) / 4;               // 12288 b128 chunks
        for (int c = t; c < chunks; c += FPS_THREADS) {
            unsigned int lla =
                (unsigned int)(size_t)(pts + (size_t)c * 4);     // LDS byte addr
            unsigned long long gga =
                (unsigned long long)(size_t)(gpts + (size_t)c * 4);
            asm volatile("global_load_async_to_lds_b128 %0, %1, off"
                         :: "v"(lla), "v"(gga) : "memory");
        }
        asm volatile("s_wait_asynccnt 0x0" ::: "memory");
    }
#else
    for (int c = t; c < NN * 3; c += FPS_THREADS) pts[c] = gpts[c];
#endif
    __syncthreads();

    // ---- per-thread registers: 16 points + running min-dists -------------
    float px[PPT], py[PPT], pz[PPT], dd[PPT];
    const int base = t * PPT;
#pragma unroll
    for (int j = 0; j < PPT; ++j) {
        px[j] = pts[(base + j) * 3 + 0];
        py[j] = pts[(base + j) * 3 + 1];
        pz[j] = pts[(base + j) * 3 + 2];
        dd[j] = 1e10f;
    }

    if (t == 0) idxs[b * KK + 0] = 0;

    const int wid  = t >> 5;
    const int lane = t & 31;
    int last = 0;

    for (int it = 1; it < KK; ++it) {
        // broadcast query point (winner of previous step) from LDS
        const float qx = pts[last * 3 + 0];
        const float qy = pts[last * 3 + 1];
        const float qz = pts[last * 3 + 2];

        // min-update + thread-local first-max argmax (no FMA contraction)
        float bv = -1.0f;
        int   bi = 0;
#pragma unroll
        for (int j = 0; j < PPT; ++j) {
            const float dx = px[j] - qx;
            const float dy = py[j] - qy;
            const float dz = pz[j] - qz;
            const float d  = __fadd_rn(__fadd_rn(__fmul_rn(dx, dx),
                                                 __fmul_rn(dy, dy)),
                                       __fmul_rn(dz, dz));
            dd[j] = fminf(dd[j], d);
            if (dd[j] > bv) { bv = dd[j]; bi = base + j; }   // strict '>' keeps first max
        }

        // wave32 shuffle reduction (prefer larger value, then smaller index)
#pragma unroll
        for (int off = 16; off > 0; off >>= 1) {
            const float ov = __shfl_down(bv, off, 32);
            const int   oi = __shfl_down(bi, off, 32);
            if (ov > bv || (ov == bv && oi < bi)) { bv = ov; bi = oi; }
        }
        if (lane == 0) { rv[wid] = bv; ri[wid] = bi; }
        __syncthreads();

        // wave 0 reduces the 32 wave winners
        if (wid == 0) {
            float v2 = rv[lane];
            int   i2 = ri[lane];
#pragma unroll
            for (int off = 16; off > 0; off >>= 1) {
                const float ov = __shfl_down(v2, off, 32);
                const int   oi = __shfl_down(i2, off, 32);
                if (ov > v2 || (ov == v2 && oi < i2)) { v2 = ov; i2 = oi; }
            }
            if (lane == 0) {
                *win = i2;
                idxs[b * KK + it] = i2;
            }
        }
        __syncthreads();
        last = *win;
    }
}

// ---------------------------------------------------------------------------
// Bilinear interpolation of BEV features at keypoints + per-(batch,cluster)
// segment max via float atomics. One block per keypoint, one thread/channel.
// Blend uses explicit non-fused mul/add to match the reference's
// ((Ia*wa + Ib*wb) + Ic*wc) + Id*wd evaluation order.
// ---------------------------------------------------------------------------
__global__ __launch_bounds__(CC)
void interp_segmax_kernel(const float* __restrict__ points,
                          const int*   __restrict__ cids,
                          const float* __restrict__ bev,
                          const int*   __restrict__ stride_p,
                          const int*   __restrict__ idxs,
                          float*       __restrict__ out) {
    const int b = blockIdx.x / KK;
    const int k = blockIdx.x % KK;
    const int c = threadIdx.x;

    const int idx = idxs[b * KK + k];
    const size_t pbase = ((size_t)b * NN + idx) * 3;
    const float pxv = points[pbase + 0];
    const float pyv = points[pbase + 1];
    const int   lbl = cids[(size_t)b * NN + idx];

    const float st = (float)stride_p[0];
    // match reference op order: (x - xmin) / voxel / stride
    const float xf = ((pxv - 0.0f)  / 0.05f) / st;
    const float yf = ((pyv + 40.0f) / 0.05f) / st;

    int x0 = (int)floorf(xf); x0 = min(max(x0, 0), WW - 1);
    int x1 = min(x0 + 1, WW - 1);
    int y0 = (int)floorf(yf); y0 = min(max(y0, 0), HH - 1);
    int y1 = min(y0 + 1, HH - 1);

    const float wa = __fmul_rn((float)x1 - xf, (float)y1 - yf);
    const float wb = __fmul_rn((float)x1 - xf, yf - (float)y0);
    const float wc = __fmul_rn(xf - (float)x0, (float)y1 - yf);
    const float wd = __fmul_rn(xf - (float)x0, yf - (float)y0);

    const float* bp = bev + ((size_t)b * CC + c) * (size_t)HH * WW;
    const float Ia = bp[y0 * WW + x0];
    const float Ib = bp[y1 * WW + x0];
    const float Ic = bp[y0 * WW + x1];
    const float Id = bp[y1 * WW + x1];

    const float v = __fadd_rn(__fadd_rn(__fadd_rn(__fmul_rn(Ia, wa),
                                                  __fmul_rn(Ib, wb)),
                                        __fmul_rn(Ic, wc)),
                              __fmul_rn(Id, wd));

    atomicMaxF(out + ((size_t)b * SS + lbl) * CC + c, v);
}

// ---------------------------------------------------------------------------
extern "C" void kernel_launch(void* const* d_in, const int* in_sizes, int n_in,
                              void* d_out, int out_size, void* d_ws, size_t ws_size,
                              hipStream_t stream) {
    const float* points   = (const float*)d_in[0];   // (B,N,3) f32
    const int*   cids     = (const int*)  d_in[1];   // (B,N)   i32
    const float* bev      = (const float*)d_in[2];   // (B,C,H,W) f32
    const int*   stride_p = (const int*)  d_in[3];   // scalar
    float*       out      = (float*)d_out;           // (B*S, C) f32
    int*         idxs     = (int*)d_ws;              // (B,K) keypoint indices

    // 1) output = -inf (segment_max identity)
    const int n = BB * SS * CC;
    init_out_kernel<<<(n + 255) / 256, 256, 0, stream>>>(out, n);

    // 2) FPS, one workgroup per batch; whole point cloud in 320KB WGP LDS
    const size_t smem = (size_t)NN * 3 * 4 + FPS_WAVES * (4 + 4) + 16;
    fps_kernel<<<BB, FPS_THREADS, smem, stream>>>(points, idxs);

    // 3) bilinear gather + segment max
    interp_segmax_kernel<<<BB * KK, CC, 0, stream>>>(points, cids, bev,
                                                     stride_p, idxs, out);
}